// Chorus_73160472920641
// MI455X (gfx1250) — compile-verified
//
#include <hip/hip_runtime.h>
#include <cstdint>
#include <math.h>

// ---------------------------------------------------------------------------
// Chorus effect, MI455X (gfx1250).
// out[b,t] = 0.5*x[b,t] + 0.25*(x[b,t-d0(t)] + x[b,t-d1(t)]),  d in [662,1102].
//
// ~340 MB HBM traffic -> ~15 us roofline at 23.3 TB/s.  Plan:
//   1. LFO delays are periodic (period 44100/1.5 = 29400): precompute the
//      2x29400 table once (exact f64 math) into d_ws; L2-resident (235 KB).
//   2. Hot kernel: TDM-stage a 16 x (TILE+HALO) tile of x into LDS with one
//      2-D D# descriptor (TENSORcnt-tracked).  Each thread owns 4 adjacent
//      columns: ds_load_b128 dry reads, scalar LDS gathers, b128 non-temporal
//      stores.  Interior blocks skip all validity selects.
// ---------------------------------------------------------------------------

typedef uint32_t u32;
typedef uint32_t __attribute__((ext_vector_type(4))) u32x4;
typedef int32_t  __attribute__((ext_vector_type(4))) i32x4;
typedef int32_t  __attribute__((ext_vector_type(8))) i32x8;
typedef float    __attribute__((ext_vector_type(4))) f32x4;

#define NB        16            // batch rows (fixed by reference)
#define TILE      1024          // time samples per block
#define HALO      1104          // > max delay (1102), multiple of 4
#define PITCH_MAX (TILE + HALO) // 2128 floats per LDS row
#define THREADS   256           // 4 columns per thread
#define PERIOD    29400         // LFO period in samples (SR / RATE)

__device__ __forceinline__ int imax(int a, int b) { return a > b ? a : b; }
__device__ __forceinline__ int imin(int a, int b) { return a < b ? a : b; }

__device__ __forceinline__ int2 lfo_delays(int t) {
    // Exact float64 math matching the numpy reference.
    const double tp = ((double)t * 1.5) / 44100.0;
    const double p0 = fmod(tp, 1.0);
    const double p1 = fmod(tp + 0.5, 1.0);
    const double TWO_PI = 6.283185307179586476925286766559;
    int d0 = 882 + (int)(sin(TWO_PI * p0) * 220.0);  // trunc toward 0
    int d1 = 882 + (int)(sin(TWO_PI * p1) * 220.0);
    d0 = imin(imax(d0, 1), 2047);
    d1 = imin(imax(d1, 1), 2047);
    return make_int2(d0, d1);
}

// ---- Kernel 1: one-period delay table (2 x 29400 ints, 235 KB) ------------
__global__ __launch_bounds__(THREADS)
void delay_table_kernel(int2* __restrict__ dtab) {
    const int t = blockIdx.x * THREADS + threadIdx.x;
    if (t < PERIOD) dtab[t] = lfo_delays(t);
}

// ---- Kernel 2: streaming chorus ------------------------------------------
template <bool USE_TABLE>
__global__ __launch_bounds__(THREADS)
void chorus_kernel(const float* __restrict__ x, float* __restrict__ out,
                   int T, const int2* __restrict__ dtab) {
    extern __shared__ float smem[];

    const int tile_start = blockIdx.x * TILE;
    const int cover_end  = imin(tile_start + TILE, T);
    const int src_col    = imax(tile_start - HALO, 0);
    const int width      = cover_end - src_col;      // <= PITCH_MAX, mult of 4
    const int pitch      = width;

    // ---- Stage x[0..15, src_col .. src_col+width) into LDS via TDM --------
#if __has_builtin(__builtin_amdgcn_tensor_load_to_lds) && \
    __has_builtin(__builtin_amdgcn_s_wait_tensorcnt)
    if (threadIdx.x < 32) {  // wave 0 issues the DMA (branch is wave-uniform)
        const uint64_t ga  = (uint64_t)(uintptr_t)(x + src_col);
        const u32      lds = (u32)(uintptr_t)(&smem[0]);

        // D# group 0 (128b): count=1, lds_addr, 57-bit global_addr, type=2
        u32x4 g0;
        g0[0] = 1u;
        g0[1] = lds;
        g0[2] = (u32)ga;
        g0[3] = (u32)((ga >> 32) & 0x01FFFFFFu) | (2u << 30);

        // D# group 1 (256b): data_size=4B; tensor/tile 16 x width; stride T.
        i32x8 g1;
        g1[0] = (int)(2u << 16);                      // data_size=4B, wg_mask=0
        g1[1] = (int)(((u32)width & 0xFFFFu) << 16);  // tensor_dim0[15:0]
        g1[2] = (int)((((u32)width >> 16) & 0xFFFFu)
              | ((u32)NB << 16));                     // dim0 hi | tensor_dim1=16
        g1[3] = (int)(((u32)width & 0xFFFFu) << 16);  // tile_dim0 = width
        g1[4] = NB;                                   // tile_dim1=16, tile_dim2=0
        g1[5] = T;                                    // tensor_dim0_stride = T
        g1[6] = 0;
        g1[7] = 0;

        i32x4 g2 = {0, 0, 0, 0};
        i32x4 g3 = {0, 0, 0, 0};
        i32x8 g4 = {0, 0, 0, 0, 0, 0, 0, 0};

        __builtin_amdgcn_tensor_load_to_lds(g0, g1, g2, g3, g4, /*cpol=*/0);
        __builtin_amdgcn_s_wait_tensorcnt(0);
    }
#else
    for (int r = 0; r < NB; ++r)
        for (int c = threadIdx.x; c < width; c += THREADS)
            smem[r * pitch + c] = x[(size_t)r * T + src_col + c];
#endif
    __syncthreads();

    // ---- Compute: 4 adjacent columns per thread ---------------------------
    const int c0 = tile_start + (int)threadIdx.x * 4;
    if (c0 >= cover_end) return;
    const int nc = imin(4, cover_end - c0);          // 4 except odd tails

    // Per-column delays (from L2-resident table or inline f64 sin).
    int gA[4], gB[4], lA[4], lB[4];
    const int m0 = c0 % PERIOD;
#pragma unroll
    for (int i = 0; i < 4; ++i) {
        int2 d;
        if (USE_TABLE) {
            int m = m0 + i;
            if (m >= PERIOD) m -= PERIOD;
            d = dtab[m];
        } else {
            d = lfo_delays(c0 + i);
        }
        gA[i] = c0 + i - d.x;
        gB[i] = c0 + i - d.y;
        lA[i] = imax(gA[i] - src_col, 0);
        lB[i] = imax(gB[i] - src_col, 0);
    }
    const int lX = c0 - src_col;                     // multiple of 4

    if (nc == 4 && tile_start >= HALO) {
        // Fast path: every gather is in-range; select-free inner loop.
        const float* row = smem;
        float* orow = out + c0;
#pragma unroll
        for (int b = 0; b < NB; ++b) {
            const f32x4 xv = *(const f32x4*)(row + lX);   // ds_load_b128
            f32x4 o;
#pragma unroll
            for (int i = 0; i < 4; ++i)
                o[i] = 0.5f * xv[i] + 0.25f * (row[lA[i]] + row[lB[i]]);
            __builtin_nontemporal_store(o, (f32x4*)orow); // b128 NT store
            row  += pitch;
            orow += T;
        }
    } else {
        // Guarded path (tile 0 warm-up zeros + any tail columns).
        const float* row = smem;
        float* orow = out + c0;
        for (int b = 0; b < NB; ++b) {
            for (int i = 0; i < nc; ++i) {
                const float xv = row[lX + i];
                const float wA = (gA[i] >= 0) ? row[lA[i]] : 0.0f;
                const float wB = (gB[i] >= 0) ? row[lB[i]] : 0.0f;
                __builtin_nontemporal_store(
                    0.5f * xv + 0.25f * (wA + wB), orow + i);
            }
            row  += pitch;
            orow += T;
        }
    }
}

extern "C" void kernel_launch(void* const* d_in, const int* in_sizes, int n_in,
                              void* d_out, int out_size, void* d_ws, size_t ws_size,
                              hipStream_t stream) {
    const float* x = (const float*)d_in[0];
    float* out     = (float*)d_out;
    const int T    = in_sizes[0] / NB;            // (B,T) flat, B = 16

    const int nTiles   = (T + TILE - 1) / TILE;
    const size_t shmem = (size_t)PITCH_MAX * NB * sizeof(float); // 136,192 B

    if (ws_size >= (size_t)PERIOD * sizeof(int2)) {
        int2* dtab = (int2*)d_ws;
        delay_table_kernel<<<dim3((PERIOD + THREADS - 1) / THREADS),
                             dim3(THREADS), 0, stream>>>(dtab);
        chorus_kernel<true><<<dim3(nTiles), dim3(THREADS), shmem, stream>>>(
            x, out, T, dtab);
    } else {
        chorus_kernel<false><<<dim3(nTiles), dim3(THREADS), shmem, stream>>>(
            x, out, T, nullptr);
    }
}